// RotRNN_12154757448447
// MI455X (gfx1250) — compile-verified
//
#include <hip/hip_runtime.h>
#include <hip/hip_bf16.h>
#include <math.h>

typedef __attribute__((ext_vector_type(2))) float v2f;
typedef __attribute__((ext_vector_type(8))) float v8f;

#define TLEN   4096
#define HID    512
#define LRU    512
#define NHEAD  64
#define NHD    8
#define NPAIR  256
#define CHUNK  128
#define NCHUNK 32

// workspace layout (float offsets)
#define OFF_W1 0
#define OFF_W2 (512*512)
#define OFF_U  (2*512*512)
#define OFF_A  (OFF_U + TLEN*512)          // per-pair decay (re,im) interleaved
#define OFF_AL (OFF_A + 2*NPAIR)           // per-pair decay^CHUNK
#define OFF_F  (OFF_AL + 2*NPAIR)          // chunk finals (NCHUNK x 512)
#define OFF_CR (OFF_F + NCHUNK*512)        // carry-in per chunk (NCHUNK x 512)
// total ~2.66M floats = ~10.2 MB of d_ws

// ---------------------------------------------------------------------------
// Setup: per head h (64 blocks):
//   E = expm(P_h - P_h^T) (8x8, Taylor series; ||A|| < ~0.6 so 24 terms >> fp32)
//   gamma, trace(B B^T), norm
//   W1[(h*8+N), d] = norm * sum_n E[N][n] * B[(h*8+n), d]      (Pm @ B_norm)
//   W2[dH, h*8+N]  = sum_n C[dH, h*8+n] * E[N][n]              (C @ blockdiag(Pm^T))
//   a  = gamma * e^{i theta}, aL = a^CHUNK  per complex pair
// ---------------------------------------------------------------------------
__global__ void k_setup(const float* __restrict__ theta_log,
                        const float* __restrict__ P,
                        const float* __restrict__ Bm,
                        const float* __restrict__ Cm,
                        const float* __restrict__ gamma_log,
                        float* __restrict__ ws)
{
    __shared__ float sE[64];
    __shared__ float sRed[64];
    const int h = blockIdx.x;
    const int tid = threadIdx.x;

    if (tid == 0) {
        float A[64], Tm[64], E[64], Tn[64];
        for (int i = 0; i < 8; ++i)
            for (int j = 0; j < 8; ++j)
                A[i*8+j] = P[(h*8+i)*8 + j] - P[(h*8+j)*8 + i];
        for (int i = 0; i < 64; ++i) { E[i] = 0.f; Tm[i] = 0.f; }
        for (int i = 0; i < 8; ++i)  { E[i*8+i] = 1.f; Tm[i*8+i] = 1.f; }
        for (int k = 1; k <= 24; ++k) {
            float invk = 1.f / (float)k;
            for (int i = 0; i < 8; ++i)
                for (int j = 0; j < 8; ++j) {
                    float s = 0.f;
                    for (int m = 0; m < 8; ++m) s += Tm[i*8+m] * A[m*8+j];
                    Tn[i*8+j] = s * invk;
                }
            for (int i = 0; i < 64; ++i) { Tm[i] = Tn[i]; E[i] += Tn[i]; }
        }
        for (int i = 0; i < 64; ++i) sE[i] = E[i];
    }

    // trace(B_h B_h^T) = sum of squares of 8x512 block
    float s = 0.f;
    for (int idx = tid; idx < NHD*HID; idx += blockDim.x) {
        float v = Bm[h*NHD*HID + idx];
        s += v * v;
    }
    sRed[tid] = s;
    __syncthreads();
    for (int off = 32; off > 0; off >>= 1) {
        if (tid < off) sRed[tid] += sRed[tid + off];
        __syncthreads();
    }
    const float trace = sRed[0];
    const float g    = expf(-expf(gamma_log[h]));
    const float norm = sqrtf((1.f - g*g) / trace);

    float* W1 = ws + OFF_W1;
    float* W2 = ws + OFF_W2;
    for (int d = tid; d < HID; d += blockDim.x) {
        for (int N = 0; N < 8; ++N) {
            float acc = 0.f;
            for (int n = 0; n < 8; ++n)
                acc += sE[N*8+n] * Bm[(h*8+n)*HID + d];
            W1[(size_t)(h*8+N)*HID + d] = norm * acc;
        }
    }
    for (int dH = tid; dH < HID; dH += blockDim.x) {
        for (int N = 0; N < 8; ++N) {
            float acc = 0.f;
            for (int n = 0; n < 8; ++n)
                acc += Cm[(size_t)dH*LRU + h*8 + n] * sE[N*8+n];
            W2[(size_t)dH*LRU + h*8 + N] = acc;
        }
    }
    if (tid < 4) {
        const int p  = h*4 + tid;
        const float th = expf(theta_log[p]);
        float* Ap  = ws + OFF_A;
        float* ApL = ws + OFF_AL;
        Ap[2*p]   = g * cosf(th);
        Ap[2*p+1] = g * sinf(th);
        const float r  = powf(g, (float)CHUNK);
        const float ph = (float)CHUNK * th;
        ApL[2*p]   = r * cosf(ph);
        ApL[2*p+1] = r * sinf(ph);
    }
}

// ---------------------------------------------------------------------------
// GEMM 1:  U[t, l] = sum_d X[d, t] * W1[l, d]        (fp32 WMMA 16x16x4)
// X is the batch-0 slice of input_sequence, stored d-major: X[d*TLEN + t].
// One wave per 16x16 tile; A: lane=M(t), VGPR v -> K = 2*(lane>>4)+v.
// ---------------------------------------------------------------------------
__global__ void __launch_bounds__(256)
k_gemm_u(const float* __restrict__ X, const float* __restrict__ W1,
         float* __restrict__ U)
{
    const int gid  = blockIdx.x * blockDim.x + threadIdx.x;
    const int wave = gid >> 5;
    const int lane = gid & 31;
    const int tm = wave >> 5;          // 256 t-tiles
    const int tn = wave & 31;          // 32 l-tiles
    const int t0 = tm * 16, l0 = tn * 16;
    const int m = lane & 15, half = lane >> 4;

    v8f acc = {};
    const float* xa = X  + (t0 + m);                  // + k*TLEN
    const float* wb = W1 + (size_t)(l0 + m) * HID;    // + k
#pragma unroll 4
    for (int d0 = 0; d0 < HID; d0 += 4) {
        const int k = d0 + 2*half;
        v2f a, b;
        a.x = xa[(size_t)(k + 0) * TLEN];
        a.y = xa[(size_t)(k + 1) * TLEN];
        b = *(const v2f*)(wb + k);
        acc = __builtin_amdgcn_wmma_f32_16x16x4_f32(
            false, a, false, b, (short)0, acc, false, false);
    }
    const int col  = l0 + m;
    const int trow = t0 + half * 8;
#pragma unroll
    for (int r = 0; r < 8; ++r)
        U[(size_t)(trow + r) * LRU + col] = acc[r];
}

// ---------------------------------------------------------------------------
// Scan phase A: per (chunk, pair) local inclusive scan, in place; record final.
// ---------------------------------------------------------------------------
__global__ void k_scan_local(float* __restrict__ U, const float* __restrict__ Ap,
                             float* __restrict__ F)
{
    const int idx   = blockIdx.x * blockDim.x + threadIdx.x;
    const int chunk = idx >> 8;
    const int c     = idx & 255;
    const float ar = Ap[2*c], ai = Ap[2*c+1];
    float zr = 0.f, zi = 0.f;
    const int base = chunk * CHUNK;
    for (int t = 0; t < CHUNK; ++t) {
        float* p = U + (size_t)(base + t) * LRU + 2*c;
        const float ur = p[0], ui = p[1];
        const float nr = ar*zr - ai*zi + ur;
        const float ni = ar*zi + ai*zr + ui;
        zr = nr; zi = ni;
        p[0] = zr; p[1] = zi;
    }
    F[(size_t)chunk * LRU + 2*c]     = zr;
    F[(size_t)chunk * LRU + 2*c + 1] = zi;
}

// Scan phase B: sequential scan over NCHUNK chunk-finals with a^CHUNK.
__global__ void k_scan_carry(const float* __restrict__ F,
                             const float* __restrict__ ApL,
                             float* __restrict__ Carry)
{
    const int c = threadIdx.x;          // 256 pairs
    const float ar = ApL[2*c], ai = ApL[2*c+1];
    float zr = 0.f, zi = 0.f;
    for (int k = 0; k < NCHUNK; ++k) {
        Carry[(size_t)k * LRU + 2*c]     = zr;
        Carry[(size_t)k * LRU + 2*c + 1] = zi;
        const float fr = F[(size_t)k * LRU + 2*c];
        const float fi = F[(size_t)k * LRU + 2*c + 1];
        const float nr = ar*zr - ai*zi + fr;
        const float ni = ar*zi + ai*zr + fi;
        zr = nr; zi = ni;
    }
}

// Scan phase C: fix-up — add a^{off+1} * carry_in to each local result.
__global__ void k_scan_fix(float* __restrict__ U, const float* __restrict__ Ap,
                           const float* __restrict__ Carry)
{
    const int idx   = blockIdx.x * blockDim.x + threadIdx.x;
    const int chunk = idx >> 8;
    const int c     = idx & 255;
    const float ar = Ap[2*c], ai = Ap[2*c+1];
    const float cr = Carry[(size_t)chunk * LRU + 2*c];
    const float ci = Carry[(size_t)chunk * LRU + 2*c + 1];
    float wr = ar*cr - ai*ci;
    float wi = ar*ci + ai*cr;
    const int base = chunk * CHUNK;
    for (int t = 0; t < CHUNK; ++t) {
        float* p = U + (size_t)(base + t) * LRU + 2*c;
        p[0] += wr;
        p[1] += wi;
        const float nr = ar*wr - ai*wi;
        const float ni = ar*wi + ai*wr;
        wr = nr; wi = ni;
    }
}

// ---------------------------------------------------------------------------
// GEMM 2 + epilogue: out[t, d] = sum_l Y[t, l] * W2[d, l] + D[d] * X[d, t]
// ---------------------------------------------------------------------------
__global__ void __launch_bounds__(256)
k_gemm_out(const float* __restrict__ Y, const float* __restrict__ W2,
           const float* __restrict__ X, const float* __restrict__ Dv,
           float* __restrict__ out)
{
    const int gid  = blockIdx.x * blockDim.x + threadIdx.x;
    const int wave = gid >> 5;
    const int lane = gid & 31;
    const int tm = wave >> 5;
    const int tn = wave & 31;
    const int t0 = tm * 16, l0 = tn * 16;
    const int m = lane & 15, half = lane >> 4;

    v8f acc = {};
    const float* ya = Y  + (size_t)(t0 + m) * LRU;    // + k
    const float* wb = W2 + (size_t)(l0 + m) * LRU;    // + k
#pragma unroll 4
    for (int k0 = 0; k0 < LRU; k0 += 4) {
        const int k = k0 + 2*half;
        v2f a = *(const v2f*)(ya + k);
        v2f b = *(const v2f*)(wb + k);
        acc = __builtin_amdgcn_wmma_f32_16x16x4_f32(
            false, a, false, b, (short)0, acc, false, false);
    }
    const int d    = l0 + m;
    const int trow = t0 + half * 8;
    const float dv = Dv[d];
#pragma unroll
    for (int r = 0; r < 8; ++r) {
        const int t = trow + r;
        out[(size_t)t * HID + d] = acc[r] + dv * X[(size_t)d * TLEN + t];
    }
}

// ---------------------------------------------------------------------------
extern "C" void kernel_launch(void* const* d_in, const int* in_sizes, int n_in,
                              void* d_out, int out_size, void* d_ws, size_t ws_size,
                              hipStream_t stream)
{
    const float* x_in      = (const float*)d_in[0]; // (8,1,512,1,4096); batch0 = first 512*4096
    const float* theta_log = (const float*)d_in[1]; // (256,1)
    const float* P         = (const float*)d_in[2]; // (512,8)
    const float* Bm        = (const float*)d_in[3]; // (512,512)
    const float* Cm        = (const float*)d_in[4]; // (512,512)
    const float* Dv        = (const float*)d_in[5]; // (512,)
    const float* gl        = (const float*)d_in[6]; // (64,)
    float* out = (float*)d_out;                     // (4096,512) fp32
    float* ws  = (float*)d_ws;

    (void)in_sizes; (void)n_in; (void)out_size; (void)ws_size;

    k_setup<<<NHEAD, 64, 0, stream>>>(theta_log, P, Bm, Cm, gl, ws);
    k_gemm_u<<<1024, 256, 0, stream>>>(x_in, ws + OFF_W1, ws + OFF_U);
    k_scan_local<<<NCHUNK, 256, 0, stream>>>(ws + OFF_U, ws + OFF_A, ws + OFF_F);
    k_scan_carry<<<1, 256, 0, stream>>>(ws + OFF_F, ws + OFF_AL, ws + OFF_CR);
    k_scan_fix<<<NCHUNK, 256, 0, stream>>>(ws + OFF_U, ws + OFF_A, ws + OFF_CR);
    k_gemm_out<<<1024, 256, 0, stream>>>(ws + OFF_U, ws + OFF_W2, x_in, Dv, out);
}